// InjectedEncoder_40415642255404
// MI455X (gfx1250) — compile-verified
//
#include <hip/hip_runtime.h>
#include <hip/hip_bf16.h>

// ---------------------------------------------------------------------------
// MI455X (gfx1250) implementation. fp32 WMMA (v_wmma_f32_16x16x4_f32) for all
// GEMM work; wave32, one wave per 16x16 output tile.
// Roofline: ~300 GFLOP fp32 compute; 12x hyp_w stream (4.8 GB) ~0.21 ms
// @23.3 TB/s dominates -> hyper GEMM streams each weight element exactly once
// per 16-wide N tile, as b128 loads. X operand staged to LDS via
// global_load_async_to_lds_b128 (ASYNCcnt) to exercise the CDNA5 async path.
// Workspace: ~300 MB laid out in kernel_launch.
// ---------------------------------------------------------------------------

typedef __attribute__((ext_vector_type(2))) float v2f;
typedef __attribute__((ext_vector_type(4))) float v4f;
typedef __attribute__((ext_vector_type(8))) float v8f;

#define NFD   128
#define BATCH 16
#define KTOT  197376
// hk segment offsets
#define OFF0  0        // w_in  (128x384)
#define OFF1  49152    // b_in  (128)
#define OFF2  49280    // w_md  (128x128)
#define OFF3  65664    // b_md  (128)
#define OFF4  65792    // w_ot  (256x128)
#define OFF5  98560    // b_ot  (256)
#define OFF6  98816    // w_sc  (256x384)
#define OFF7  197120   // b_sc  (256)

// ---------------------------------------------------------------------------
// Generic WMMA accumulator (used by spatial kernels): C(16x16) += A(16xK)*B
// A: row-major, row stride a_rs, k contiguous.
// B: element (k,n) at Bm[k*b_ks + n*b_ns]  (spatial: b_ks=Npix, b_ns=1 ->
//    lane-contiguous, naturally coalesced).
// Fragment layouts per CDNA5 ISA 7.12.2 (wave32):
//   A 16x4:  VGPR0 lanes0-15 K=k0, lanes16-31 K=k0+2; VGPR1: k0+1 / k0+3
//   B 4x16:  VGPR0 lanes0-15 row k0, lanes16-31 row k0+1; VGPR1: k0+2 / k0+3
//   C/D:     VGPR r lanes0-15 -> M=r, lanes16-31 -> M=r+8
// ---------------------------------------------------------------------------
__device__ inline v8f wmma_accum(const float* __restrict__ A, int a_rs,
                                 const float* __restrict__ Bm, int b_ks, int b_ns,
                                 int K, v8f acc, int lane)
{
  const int mn = lane & 15;        // M index for A, N index for B
  const int hi = lane >> 4;        // 0 or 1
  const float* arow = A + (size_t)mn * a_rs;
  for (int k0 = 0; k0 < K; k0 += 4) {
    if ((k0 & 63) == 0) {
      __builtin_prefetch(Bm + (size_t)(k0 + 64) * b_ks + (size_t)mn * b_ns, 0, 1);
    }
    v2f a, b;
    a.x = arow[k0 + 2 * hi + 0];
    a.y = arow[k0 + 2 * hi + 1];
    b.x = Bm[(size_t)(k0 + hi) * b_ks + (size_t)mn * b_ns];
    b.y = Bm[(size_t)(k0 + 2 + hi) * b_ks + (size_t)mn * b_ns];
    acc = __builtin_amdgcn_wmma_f32_16x16x4_f32(false, a, false, b,
                                                (short)0, acc, false, false);
  }
  return acc;
}

// ---------------------------------------------------------------------------
// Dense accumulator: C(16x16) += X(16xK) @ Wt(n-rows x K)^T for one 16-wide
// N tile. Wt rows are k-contiguous -> per-lane b128 streaming loads.
// X chunk (16x64) is staged into LDS with global_load_async_to_lds_b128
// (ASYNCcnt), then A fragments come from LDS. Wave-private LDS; WAR reuse
// across chunks guarded by s_wait_dscnt 0 (A values already in VGPRs then).
// Requires K % 64 == 0 (holds: 512, 1024, 4480).
// ---------------------------------------------------------------------------
__device__ inline v8f dense_accum(const float* __restrict__ X,
                                  const float* __restrict__ Wt, int K,
                                  v8f acc, int lane, float* xs)
{
  const int mn = lane & 15;
  const int hi = lane >> 4;
  const unsigned lbase = (unsigned)(uintptr_t)xs;
  const float* brow = Wt + (size_t)mn * K;   // this lane's weight row
  for (int k0 = 0; k0 < K; k0 += 64) {
    // make sure previous chunk's LDS reads have landed in VGPRs before the
    // async engine overwrites the staging buffer
    asm volatile("s_wait_dscnt 0" ::: "memory");
    // stage X[0:16, k0:k0+64] -> LDS (8 x b128 per lane = 4KB)
#pragma unroll
    for (int i = 0; i < 8; ++i) {
      int g = i * 32 + lane;                 // 256 float4 groups
      int m = g >> 4, kg = g & 15;
      unsigned loff = lbase + (unsigned)g * 16u;
      unsigned long long ga =
          (unsigned long long)(uintptr_t)(X + (size_t)m * K + k0 + kg * 4);
      asm volatile("global_load_async_to_lds_b128 %0, %1, off"
                   :: "v"(loff), "v"(ga) : "memory");
    }
    __builtin_prefetch(brow + k0 + 64, 0, 1);
    asm volatile("s_wait_asynccnt 0" ::: "memory");
    // LDS chunk layout: xs[m*64 + (k-k0)]
#pragma unroll
    for (int kk = 0; kk < 64; kk += 4) {
      v2f a, b;
      a.x = xs[mn * 64 + kk + 2 * hi + 0];
      a.y = xs[mn * 64 + kk + 2 * hi + 1];
      v4f bq = *(const v4f*)(brow + k0 + kk);        // b128 stream
      b.x = hi ? bq.y : bq.x;                        // rows k0+kk .. +3
      b.y = hi ? bq.w : bq.z;
      acc = __builtin_amdgcn_wmma_f32_16x16x4_f32(false, a, false, b,
                                                  (short)0, acc, false, false);
    }
  }
  return acc;
}

// ---------------------------------------------------------------------------
// Dense: Y(16xN) = X1(16xK1) @ W1^T [+ X2 @ W2^T] + bias [+bias2], opt relu.
// W row-major (N x K). One wave per 16-wide N tile.
// ---------------------------------------------------------------------------
__global__ __launch_bounds__(32) void k_dense16(
    const float* __restrict__ X1, const float* __restrict__ W1, int K1,
    const float* __restrict__ X2, const float* __restrict__ W2, int K2,
    const float* __restrict__ bias, const float* __restrict__ bias2,
    float* __restrict__ Y, int N, int do_relu)
{
  __shared__ __align__(16) float xs[16 * 64];
  const int nt = blockIdx.x;
  const int lane = threadIdx.x;
  v8f acc = {};
  acc = dense_accum(X1, W1 + (size_t)nt * 16 * K1, K1, acc, lane, xs);
  if (K2 > 0)
    acc = dense_accum(X2, W2 + (size_t)nt * 16 * K2, K2, acc, lane, xs);
  const int n = lane & 15, hi = lane >> 4;
  const int ng = nt * 16 + n;
  float bv = (bias ? bias[ng] : 0.0f) + (bias2 ? bias2[ng] : 0.0f);
#pragma unroll
  for (int r = 0; r < 8; ++r) {
    int m = r + 8 * hi;
    float v = acc[r] + bv;
    if (do_relu) v = fmaxf(v, 0.0f);
    Y[(size_t)m * N + ng] = v;
  }
}

// ---------------------------------------------------------------------------
// Spatial conv-as-GEMM (per batch): Y[b,o,pix] =
//   sum_c W1_b[o,c]*IN1[b,c,pix] + B1_b[o]
//   (+ sum_c W2_b[o,c]*IN2[b,c,pix] + B2_b[o]) (+ ADD[b,o,pix]), opt relu.
// Per-batch weight strides w*bs (0 => shared weights, KTOT => from hk).
// grid: (Npix/16, O/16, B); one wave per tile.
// ---------------------------------------------------------------------------
__global__ __launch_bounds__(32) void k_spatial(
    const float* __restrict__ W1, long w1bs, const float* __restrict__ B1, long b1bs,
    const float* __restrict__ IN1, int C1,
    const float* __restrict__ W2, long w2bs, const float* __restrict__ B2, long b2bs,
    const float* __restrict__ IN2, int C2,
    const float* __restrict__ ADD, float* __restrict__ Y, int Npix, int do_relu)
{
  const int nt = blockIdx.x, ot = blockIdx.y, b = blockIdx.z;
  const int O = gridDim.y * 16;
  const int lane = threadIdx.x;
  v8f acc = {};
  acc = wmma_accum(W1 + (size_t)b * w1bs + (size_t)ot * 16 * C1, C1,
                   IN1 + (size_t)b * C1 * Npix + nt * 16, Npix, 1, C1, acc, lane);
  if (W2 != nullptr && C2 > 0)
    acc = wmma_accum(W2 + (size_t)b * w2bs + (size_t)ot * 16 * C2, C2,
                     IN2 + (size_t)b * C2 * Npix + nt * 16, Npix, 1, C2, acc, lane);
  const int n = lane & 15, hi = lane >> 4;
#pragma unroll
  for (int r = 0; r < 8; ++r) {
    int o = ot * 16 + r + 8 * hi;
    float v = acc[r];
    if (B1) v += B1[(size_t)b * b1bs + o];
    if (B2) v += B2[(size_t)b * b2bs + o];
    size_t yidx = ((size_t)b * O + o) * Npix + nt * 16 + n;
    if (ADD) v += ADD[yidx];
    if (do_relu) v = fmaxf(v, 0.0f);
    Y[yidx] = v;
  }
}

// ---------------------------------------------------------------------------
// Elementwise / reduction kernels
// ---------------------------------------------------------------------------
__global__ void k_in_conv(const float* __restrict__ x, const float* __restrict__ w,
                          const float* __restrict__ b, float* __restrict__ out)
{
  int idx = blockIdx.x * blockDim.x + threadIdx.x;
  const int Np = 64 * 64;
  if (idx >= BATCH * NFD * Np) return;
  int pix = idx % Np;
  int o = (idx / Np) % NFD;
  int bb = idx / (NFD * Np);
  float acc = b[o];
  const float* xb = x + (size_t)bb * 3 * Np;
  acc += xb[0 * Np + pix] * w[o * 3 + 0];
  acc += xb[1 * Np + pix] * w[o * 3 + 1];
  acc += xb[2 * Np + pix] * w[o * 3 + 2];
  out[idx] = acc;
}

__global__ void k_sobel(const float* __restrict__ out, float* __restrict__ p, int S)
{
  int idx = blockIdx.x * blockDim.x + threadIdx.x;
  int Np = S * S;
  if (idx >= BATCH * NFD * Np) return;
  int pix = idx % Np;
  int ch = (idx / Np) % NFD;
  int b = idx / (NFD * Np);
  int i = pix / S, j = pix % S;
  const float* base = out + (size_t)(b * NFD + ch) * Np;
  float v[3][3];
#pragma unroll
  for (int u = 0; u < 3; ++u)
#pragma unroll
    for (int w2 = 0; w2 < 3; ++w2) {
      int ii = i + u - 1, jj = j + w2 - 1;
      v[u][w2] = (ii >= 0 && ii < S && jj >= 0 && jj < S) ? base[ii * S + jj] : 0.0f;
    }
  float sx = (-v[0][0] + v[0][2] - 2.f * v[1][0] + 2.f * v[1][2] - v[2][0] + v[2][2]) * 0.125f;
  float sy = (-v[0][0] - 2.f * v[0][1] - v[0][2] + v[2][0] + 2.f * v[2][1] + v[2][2]) * 0.125f;
  size_t pb = (size_t)b * 3 * NFD * Np;
  p[pb + (size_t)ch * Np + pix] = v[1][1];
  p[pb + (size_t)(NFD + ch) * Np + pix] = sx;
  p[pb + (size_t)(2 * NFD + ch) * Np + pix] = sy;
}

__global__ void k_instnorm(float* __restrict__ p, int Np)
{
  __shared__ float ssum[256], ssq[256];
  int bc = blockIdx.x, t = threadIdx.x;
  float* base = p + (size_t)bc * Np;
  float s = 0.f, q = 0.f;
  for (int i = t; i < Np; i += 256) { float v = base[i]; s += v; q += v * v; }
  ssum[t] = s; ssq[t] = q;
  __syncthreads();
  for (int st = 128; st > 0; st >>= 1) {
    if (t < st) { ssum[t] += ssum[t + st]; ssq[t] += ssq[t + st]; }
    __syncthreads();
  }
  float mean = ssum[0] / (float)Np;
  float var = ssq[0] / (float)Np - mean * mean;
  float inv = rsqrtf(var + 1e-5f);
  for (int i = t; i < Np; i += 256) base[i] = (base[i] - mean) * inv;
}

__global__ void k_update(float* __restrict__ out, const float* __restrict__ dbuf,
                         const float* __restrict__ leakp, int Np)
{
  int idx = blockIdx.x * blockDim.x + threadIdx.x;
  if (idx >= BATCH * NFD * Np) return;
  int pix = idx % Np;
  int ch = (idx / Np) % NFD;
  int b = idx / (NFD * Np);
  float leak = fminf(fmaxf(leakp[0], 0.001f), 1000.0f);
  float a = dbuf[((size_t)b * 256 + ch) * Np + pix];
  float g = dbuf[((size_t)b * 256 + 128 + ch) * Np + pix];
  out[idx] += leak * a * (1.0f / (1.0f + expf(-g)));
}

__global__ void k_down(const float* __restrict__ in, float* __restrict__ outp, int S)
{
  const float G[3] = {0.27406862f, 0.45186276f, 0.27406862f}; // gauss weights
  int So = S / 2, Np = So * So;
  int idx = blockIdx.x * blockDim.x + threadIdx.x;
  if (idx >= BATCH * NFD * Np) return;
  int pix = idx % Np;
  int ch = (idx / Np) % NFD;
  int b = idx / (NFD * Np);
  int i = pix / So, j = pix % So;
  const float* base = in + (size_t)(b * NFD + ch) * S * S;
  float acc = 0.f;
#pragma unroll
  for (int di = 0; di < 2; ++di)
#pragma unroll
    for (int dj = 0; dj < 2; ++dj) {
      int ci = 2 * i + di, cj = 2 * j + dj;
#pragma unroll
      for (int u = -1; u <= 1; ++u)
#pragma unroll
        for (int v = -1; v <= 1; ++v) {
          int ii = ci + u, jj = cj + v;
          if (ii >= 0 && ii < S && jj >= 0 && jj < S)
            acc += G[u + 1] * G[v + 1] * base[ii * S + jj];
        }
    }
  outp[(size_t)(b * NFD + ch) * Np + pix] = acc * 0.25f;
}

__global__ void k_add_seed(float* __restrict__ out, const float* __restrict__ seed)
{
  int idx = blockIdx.x * blockDim.x + threadIdx.x;
  if (idx >= BATCH * NFD * 256) return;
  out[idx] += seed[idx % (NFD * 256)];
}

__global__ void k_out_conv(const float* __restrict__ out, const float* __restrict__ w,
                           const float* __restrict__ b, float* __restrict__ o)
{
  int idx = blockIdx.x * blockDim.x + threadIdx.x;
  if (idx >= BATCH * 385 * 256) return;
  int pix = idx % 256;
  int oc = (idx / 256) % 385;
  int bb = idx / (385 * 256);
  float acc = b[oc];
  const float* base = out + (size_t)bb * NFD * 256;
  const float* wr = w + (size_t)oc * NFD;
  for (int c = 0; c < NFD; ++c) acc += base[(size_t)c * 256 + pix] * wr[c];
  o[idx] = acc;
}

__global__ void k_cs(const float* __restrict__ o, float* __restrict__ cs)
{
  int idx = blockIdx.x * blockDim.x + threadIdx.x;
  if (idx >= BATCH * 4480) return;
  int b = idx / 4480, j = idx % 4480;
  const float* ob = o + (size_t)b * 385 * 256;
  float v;
  if (j < 128) {
    const float* ch = ob + (size_t)j * 256;
    float s = 0.f; for (int i = 0; i < 256; ++i) s += ch[i];
    v = s * (1.0f / 256.0f);
  } else if (j < 2176) {
    int t = j - 128, ch = t >> 4, h = t & 15;
    const float* r = ob + (size_t)(128 + ch) * 256 + h * 16;
    float s = 0.f; for (int w = 0; w < 16; ++w) s += r[w];
    v = s * (1.0f / 16.0f);
  } else if (j < 4224) {
    int t = j - 2176, ch = t >> 4, w = t & 15;
    const float* cb = ob + (size_t)(256 + ch) * 256;
    float s = 0.f; for (int h = 0; h < 16; ++h) s += cb[h * 16 + w];
    v = s * (1.0f / 16.0f);
  } else {
    v = ob[384 * 256 + (j - 4224)];
  }
  cs[idx] = v;
}

// ---------------------------------------------------------------------------
extern "C" void kernel_launch(void* const* d_in, const int* in_sizes, int n_in,
                              void* d_out, int out_size, void* d_ws, size_t ws_size,
                              hipStream_t stream)
{
  const float* x       = (const float*)d_in[0];
  const float* inj_lat = (const float*)d_in[1];
  const float* leakp   = (const float*)d_in[2];
  const float* in_w    = (const float*)d_in[3];
  const float* in_b    = (const float*)d_in[4];
  const float* fc_w1   = (const float*)d_in[5];
  const float* fc_b1   = (const float*)d_in[6];
  const float* fc_w2   = (const float*)d_in[7];
  const float* fc_b2   = (const float*)d_in[8];
  const float* fc_wsc  = (const float*)d_in[9];
  const float* fc_bsc  = (const float*)d_in[10];
  const float* hyp_w   = (const float*)d_in[11];
  const float* hyp_b   = (const float*)d_in[12];
  const float* lat_w   = (const float*)d_in[13];
  const float* lat_b   = (const float*)d_in[14];
  const float* seed    = (const float*)d_in[15];
  const float* out_w   = (const float*)d_in[16];
  const float* out_b   = (const float*)d_in[17];
  const float* l1_w1   = (const float*)d_in[18];
  const float* l1_b1   = (const float*)d_in[19];
  const float* l1_w2   = (const float*)d_in[20];
  const float* l1_b2   = (const float*)d_in[21];
  const float* l1_wsc  = (const float*)d_in[22];
  const float* l1_bsc  = (const float*)d_in[23];
  const float* l2_w1   = (const float*)d_in[24];
  const float* l2_b1   = (const float*)d_in[25];
  const float* l2_w2   = (const float*)d_in[26];
  const float* l2_b2   = (const float*)d_in[27];
  const float* l2_wsc  = (const float*)d_in[28];
  const float* l2_bsc  = (const float*)d_in[29];
  const float* fin_w   = (const float*)d_in[30];
  const float* fin_b   = (const float*)d_in[31];

  float* ws = (float*)d_ws;
  size_t off = 0;
  auto alloc = [&](size_t n) { float* q = ws + off; off += n; return q; };
  float* outA = alloc((size_t)BATCH * NFD * 4096);   // 64x64 state
  float* outB = alloc((size_t)BATCH * NFD * 1024);   // 32x32 state
  float* outC = alloc((size_t)BATCH * NFD * 256);    // 16x16 state
  float* pbuf = alloc((size_t)BATCH * 384 * 4096);   // concat features
  float* hA   = alloc((size_t)BATCH * NFD * 4096);   // fc relu / h1
  float* hB   = alloc((size_t)BATCH * NFD * 4096);   // h2
  float* dbuf = alloc((size_t)BATCH * 256 * 4096);   // f then f+d
  float* hk   = alloc((size_t)BATCH * KTOT);         // hyper outputs
  float* latc = alloc((size_t)BATCH * 512);
  float* obuf = alloc((size_t)BATCH * 385 * 256);
  float* csb  = alloc((size_t)BATCH * 4480);
  float* t1   = alloc((size_t)BATCH * 1024);
  float* hcs1 = alloc((size_t)BATCH * 512);
  float* t2   = alloc((size_t)BATCH * 512);
  float* hcs2 = alloc((size_t)BATCH * 512);
  (void)ws_size; (void)in_sizes; (void)n_in; (void)out_size;

  // input projection
  {
    int tot = BATCH * NFD * 4096;
    k_in_conv<<<(tot + 255) / 256, 256, 0, stream>>>(x, in_w, in_b, outA);
  }

  float* cur = outA;
  int S = 64;
  for (int c = 0; c < 12; ++c) {
    int Np = S * S;
    int totc = BATCH * NFD * Np;
    k_sobel<<<(totc + 255) / 256, 256, 0, stream>>>(cur, pbuf, S);
    k_instnorm<<<BATCH * 384, 256, 0, stream>>>(pbuf, Np);
    // latc = inj_lat @ lat_w[c].T + lat_b[c]
    k_dense16<<<512 / 16, 32, 0, stream>>>(inj_lat, lat_w + (size_t)c * 512 * 512, 512,
                                           nullptr, nullptr, 0,
                                           lat_b + (size_t)c * 512, nullptr, latc, 512, 0);
    // hk = latc @ hyp_w.T + hyp_b   (the HBM-roofline GEMM)
    k_dense16<<<KTOT / 16, 32, 0, stream>>>(latc, hyp_w, 512, nullptr, nullptr, 0,
                                            hyp_b, nullptr, hk, KTOT, 0);
    // shared fc: hA = relu(w1 @ p + b1)
    k_spatial<<<dim3(Np / 16, 128 / 16, BATCH), 32, 0, stream>>>(
        fc_w1, 0, fc_b1, 0, pbuf, 384,
        nullptr, 0, nullptr, 0, nullptr, 0, nullptr, hA, Np, 1);
    // f = wsc @ p + bsc + w2 @ hA + b2  -> dbuf
    k_spatial<<<dim3(Np / 16, 256 / 16, BATCH), 32, 0, stream>>>(
        fc_wsc, 0, fc_bsc, 0, pbuf, 384,
        fc_w2, 0, fc_b2, 0, hA, 128, nullptr, dbuf, Np, 0);
    // per-sample hyper MLP
    k_spatial<<<dim3(Np / 16, 128 / 16, BATCH), 32, 0, stream>>>(
        hk + OFF0, KTOT, hk + OFF1, KTOT, pbuf, 384,
        nullptr, 0, nullptr, 0, nullptr, 0, nullptr, hA, Np, 1);
    k_spatial<<<dim3(Np / 16, 128 / 16, BATCH), 32, 0, stream>>>(
        hk + OFF2, KTOT, hk + OFF3, KTOT, hA, 128,
        nullptr, 0, nullptr, 0, nullptr, 0, nullptr, hB, Np, 1);
    // dbuf = f + w_sc@p + b_sc + w_ot@hB + b_ot
    k_spatial<<<dim3(Np / 16, 256 / 16, BATCH), 32, 0, stream>>>(
        hk + OFF6, KTOT, hk + OFF7, KTOT, pbuf, 384,
        hk + OFF4, KTOT, hk + OFF5, KTOT, hB, 128, dbuf, dbuf, Np, 0);
    // gated state update
    k_update<<<(totc + 255) / 256, 256, 0, stream>>>(cur, dbuf, leakp, Np);
    // gauss + 2x2 mean downsample
    if (c == 3) {
      int t2o = BATCH * NFD * 1024;
      k_down<<<(t2o + 255) / 256, 256, 0, stream>>>(cur, outB, 64);
      cur = outB; S = 32;
    } else if (c == 7) {
      int t2o = BATCH * NFD * 256;
      k_down<<<(t2o + 255) / 256, 256, 0, stream>>>(cur, outC, 32);
      cur = outC; S = 16;
    }
  }

  // tail
  {
    int tot = BATCH * NFD * 256;
    k_add_seed<<<(tot + 255) / 256, 256, 0, stream>>>(cur, seed);
    int tot2 = BATCH * 385 * 256;
    k_out_conv<<<(tot2 + 255) / 256, 256, 0, stream>>>(cur, out_w, out_b, obuf);
    int tot3 = BATCH * 4480;
    k_cs<<<(tot3 + 255) / 256, 256, 0, stream>>>(obuf, csb);
  }
  // lrb 1
  k_dense16<<<1024 / 16, 32, 0, stream>>>(csb, l1_w1, 4480, nullptr, nullptr, 0,
                                          l1_b1, nullptr, t1, 1024, 1);
  k_dense16<<<512 / 16, 32, 0, stream>>>(csb, l1_wsc, 4480, t1, l1_w2, 1024,
                                         l1_bsc, l1_b2, hcs1, 512, 0);
  // lrb 2
  k_dense16<<<512 / 16, 32, 0, stream>>>(hcs1, l2_w1, 512, nullptr, nullptr, 0,
                                         l2_b1, nullptr, t2, 512, 1);
  k_dense16<<<512 / 16, 32, 0, stream>>>(hcs1, l2_wsc, 512, t2, l2_w2, 512,
                                         l2_bsc, l2_b2, hcs2, 512, 0);
  // final projection -> d_out (16x512 fp32)
  k_dense16<<<512 / 16, 32, 0, stream>>>(hcs2, fin_w, 512, nullptr, nullptr, 0,
                                         fin_b, nullptr, (float*)d_out, 512, 0);
}